// MaxReLU_78099685310945
// MI455X (gfx1250) — compile-verified
//
#include <hip/hip_runtime.h>

#define TILE  32
#define HALO  1
#define LDSW  (TILE + 2 * HALO)   // 34
#define IMG_H 224
#define IMG_W 224
#define SLOPE 0.01f

// Use the gfx1250 async global->LDS path when the toolchain exposes it.
#if defined(__HIP_DEVICE_COMPILE__) && defined(__gfx1250__) && \
    __has_builtin(__builtin_amdgcn_global_load_async_to_lds_b32)
#define USE_ASYNC_LDS 1
#else
#define USE_ASYNC_LDS 0
#endif

typedef __attribute__((address_space(1))) int* gptr_i32;
typedef __attribute__((address_space(3))) int* lptr_i32;

__global__ __launch_bounds__(256)
void maxrelu_tile_kernel(const float* __restrict__ x, float* __restrict__ out) {
    __shared__ float tile[LDSW * LDSW];   // 34*34*4 = 4624 bytes

    const int tid   = threadIdx.x;        // 0..255
    const int bx    = blockIdx.x;         // 0..6
    const int by    = blockIdx.y;         // 0..6
    const int plane = blockIdx.z;         // 0..B*C-1

    const float* __restrict__ src = x   + (size_t)plane * (IMG_H * IMG_W);
    float*       __restrict__ dst = out + (size_t)plane * (IMG_H * IMG_W);

    const int gx0 = bx * TILE - HALO;     // top-left of LDS tile in (possibly
    const int gy0 = by * TILE - HALO;     //  negative) image coordinates

    // ---- Stage 34x34 reflect-padded tile into LDS (async DMA path) ----
    // 34*34 = 1156 elements; 256 threads -> 5 straight-line batches, all
    // async loads in flight before the single asynccnt wait.
#pragma unroll
    for (int b = 0; b < (LDSW * LDSW + 255) / 256; ++b) {
        const int i = tid + b * 256;
        if (i < LDSW * LDSW) {
            const int ly = i / LDSW;
            const int lx = i - ly * LDSW;
            int gy = gy0 + ly;
            int gx = gx0 + lx;
            gy = (gy < 0) ? -gy : gy;                               // reflect
            gy = (gy > IMG_H - 1) ? (2 * (IMG_H - 1) - gy) : gy;
            gx = (gx < 0) ? -gx : gx;
            gx = (gx > IMG_W - 1) ? (2 * (IMG_W - 1) - gx) : gx;
            const float* gp = src + gy * IMG_W + gx;
#if USE_ASYNC_LDS
            __builtin_amdgcn_global_load_async_to_lds_b32(
                (gptr_i32)gp, (lptr_i32)&tile[i], /*offset=*/0, /*cpol=*/0);
#else
            tile[i] = *gp;
#endif
        }
    }

#if USE_ASYNC_LDS
#if __has_builtin(__builtin_amdgcn_s_wait_asynccnt)
    __builtin_amdgcn_s_wait_asynccnt(0);
#else
    asm volatile("s_wait_asynccnt 0" ::: "memory");
#endif
#endif
    __syncthreads();

    // ---- Compute: each thread owns one column, 4 consecutive output rows ----
    const int c  = tid & (TILE - 1);      // output column within tile, 0..31
    const int ty = tid >> 5;              // 0..7
    const int r0 = ty * 4;                // first of 4 output rows

    // Row partials over 6 LDS rows (vertical sliding-window reuse).
    float hmax[6], hsum[6], hssq[6];
#pragma unroll
    for (int k = 0; k < 6; ++k) {
        const float a = tile[(r0 + k) * LDSW + c + 0];
        const float b = tile[(r0 + k) * LDSW + c + 1];
        const float d = tile[(r0 + k) * LDSW + c + 2];
        hmax[k] = fmaxf(a, fmaxf(b, d));
        hsum[k] = a + b + d;
        hssq[k] = fmaf(a, a, fmaf(b, b, d * d));
    }

    const float inv9 = 1.0f / 9.0f;
    const int   oy0  = by * TILE;
    const int   ox   = bx * TILE + c;
#pragma unroll
    for (int k = 0; k < 4; ++k) {
        const int r  = r0 + k;
        const float M  = fmaxf(hmax[k], fmaxf(hmax[k + 1], hmax[k + 2]));
        const float s  = hsum[k] + hsum[k + 1] + hsum[k + 2];
        const float ss = hssq[k] + hssq[k + 1] + hssq[k + 2];
        const float mean = s * inv9;
        float var = fmaf(-mean, mean, ss * inv9);
        var = fmaxf(var, 0.0f);
        const float err = __builtin_sqrtf(var) * 0.1f;
        const float mid = tile[(r + 1) * LDSW + (c + 1)];
        const float res = (mid >= M - err) ? M : (SLOPE * mid);
        // Output is write-once / never re-read: keep it out of L2 so the
        // 192MB L2 stays warm with input lines (halo reuse across tiles).
        __builtin_nontemporal_store(res, &dst[(size_t)(oy0 + r) * IMG_W + ox]);
    }
}

extern "C" void kernel_launch(void* const* d_in, const int* in_sizes, int n_in,
                              void* d_out, int out_size, void* d_ws, size_t ws_size,
                              hipStream_t stream) {
    const float* x   = (const float*)d_in[0];
    float*       out = (float*)d_out;

    const int planes = in_sizes[0] / (IMG_H * IMG_W);   // B*C = 1024
    dim3 grid(IMG_W / TILE, IMG_H / TILE, planes);      // 7 x 7 x 1024
    dim3 block(256);
    maxrelu_tile_kernel<<<grid, block, 0, stream>>>(x, out);
}